// GATv3Convolution_72911364817016
// MI455X (gfx1250) — compile-verified
//
#include <hip/hip_runtime.h>
#include <hip/hip_bf16.h>
#include <math.h>

// GATv2 x2 on gfx1250. fp32 WMMA (V_WMMA_F32_16X16X4_F32) for the 6 dense
// GEMMs; L2-resident gather/scatter edge phase with f32 atomics.
// N=50000, E=800000 (+N self loops), F_IN=256, H=64, C=40, slope=0.2.

typedef __attribute__((ext_vector_type(2))) float v2f;
typedef __attribute__((ext_vector_type(8))) float v8f;

struct GemmTask {
  const float* B;      // [K, NOUT] row-major
  const float* biasA;  // [NOUT]
  const float* biasB;  // optional second bias (nullptr if none)
  float*       out;    // [nrows, NOUT] row-major
};

__device__ __forceinline__ void atomic_max_f32(float* addr, float v) {
  // sign-split trick: IEEE ordering matches int ordering for v>=0 and
  // reversed uint ordering for v<0. emax initialized to -inf.
  if (v >= 0.0f) atomicMax((int*)addr, __float_as_int(v));
  else           atomicMin((unsigned int*)addr, __float_as_uint(v));
}

// ---------------------------------------------------------------------------
// Fused triple GEMM: out = A @ B + bias. blockIdx.y in {0,1,2} picks the task.
// 256 threads = 8 waves; wave w computes rows [r0, r0+16) x NTILES*16 cols.
// ---------------------------------------------------------------------------
template<int KDIM, int NOUT, int NTILES>
__global__ void gemm_wmma_kernel(const float* __restrict__ A, int nrows,
                                 GemmTask t0, GemmTask t1, GemmTask t2) {
  GemmTask t = (blockIdx.y == 0) ? t0 : ((blockIdx.y == 1) ? t1 : t2);
  const int lane = threadIdx.x & 31;
  const int wave = threadIdx.x >> 5;
  const int r0 = blockIdx.x * 128 + wave * 16;
  if (r0 >= nrows) return;                      // wave-uniform exit

  // A fragment addressing (16x4 f32): lanes 0-15 hold K={k,k+1} of row m=lane,
  // lanes 16-31 hold K={k+2,k+3} of row m=lane-16.
  const int m  = lane & 15;
  const int kk = (lane >> 4) << 1;              // 0 or 2
  int arow_i = r0 + m; if (arow_i > nrows - 1) arow_i = nrows - 1;  // clamp, keep EXEC full
  const float* arow = A + (size_t)arow_i * KDIM;

  v8f acc[NTILES] = {};

  for (int k = 0; k < KDIM; k += 4) {
    v2f a = *(const v2f*)(arow + k + kk);       // 8B-aligned (kk even)
    #pragma unroll
    for (int ti = 0; ti < NTILES; ++ti) {
      const int n = ti * 16 + (lane & 15);
      float b0 = 0.0f, b1 = 0.0f;
      if ((NOUT % 16 == 0) || (n < NOUT)) {     // pad cols with zero for NOUT=40
        b0 = t.B[(size_t)(k + kk)     * NOUT + n];
        b1 = t.B[(size_t)(k + kk + 1) * NOUT + n];
      }
      v2f b = {b0, b1};
      acc[ti] = __builtin_amdgcn_wmma_f32_16x16x4_f32(
          false, a, false, b, (short)0, acc[ti], false, false);
    }
  }

  // C/D layout: element acc[ti][v] is (row = v + (lane>=16 ? 8:0), col = lane&15)
  const int radd = (lane >> 4) << 3;
  const bool fullrows = (r0 + 16 <= nrows);     // wave-uniform fast path
  const bool haveB    = (t.biasB != nullptr);   // uniform
  #pragma unroll
  for (int ti = 0; ti < NTILES; ++ti) {
    const int n = ti * 16 + (lane & 15);
    float bias = 0.0f;
    if ((NOUT % 16 == 0) || (n < NOUT)) {
      bias = t.biasA[n];
      if (haveB) bias += t.biasB[n];
    }
    float* op = t.out + (size_t)(r0 + radd) * NOUT + n;
    if (fullrows) {
      if ((NOUT % 16 == 0) || (n < NOUT)) {
        #pragma unroll
        for (int v = 0; v < 8; ++v)
          op[(size_t)v * NOUT] = acc[ti][v] + bias;
      }
    } else {
      #pragma unroll
      for (int v = 0; v < 8; ++v) {
        if (((NOUT % 16 == 0) || (n < NOUT)) && (r0 + radd + v < nrows))
          op[(size_t)v * NOUT] = acc[ti][v] + bias;
      }
    }
  }
}

// ---------------------------------------------------------------------------
// Segment softmax scaffolding
// ---------------------------------------------------------------------------
__global__ void seg_init_kernel(float* emax, float* denom, int n) {
  int i = blockIdx.x * blockDim.x + threadIdx.x;
  if (i < n) { emax[i] = -__builtin_inff(); denom[i] = 0.0f; }
}

// wave per edge: e = leaky_relu(xl[src]+xr[dst]) . att ; segment max over dst
template<int F>
__global__ void edge_logit_kernel(const float* __restrict__ xl,
                                  const float* __restrict__ xr,
                                  const float* __restrict__ att,
                                  const long long* __restrict__ ei,
                                  int Ee, int EL,
                                  float* __restrict__ elog,
                                  float* __restrict__ emax) {
  const int wid  = blockIdx.x * (blockDim.x >> 5) + (threadIdx.x >> 5);
  const int lane = threadIdx.x & 31;
  if (wid >= EL) return;                        // wave-uniform
  long long s, d;
  if (wid < Ee) { s = ei[wid]; d = ei[(size_t)Ee + wid]; }
  else          { s = wid - Ee; d = s; }        // self loop
  float part = 0.0f;
  for (int f = lane; f < F; f += 32) {
    float v = xl[(size_t)s * F + f] + xr[(size_t)d * F + f];
    v = (v >= 0.0f) ? v : 0.2f * v;             // leaky relu, slope 0.2
    part += v * att[f];
  }
  #pragma unroll
  for (int off = 16; off > 0; off >>= 1) part += __shfl_down(part, off, 32);
  if (lane == 0) {
    elog[wid] = part;
    atomic_max_f32(&emax[d], part);
  }
}

// thread per edge: elog <- exp(e - emax[dst]) ; denom[dst] += elog
__global__ void edge_exp_kernel(const long long* __restrict__ ei, int Ee, int EL,
                                float* __restrict__ elog,
                                const float* __restrict__ emax,
                                float* __restrict__ denom) {
  const int e = blockIdx.x * blockDim.x + threadIdx.x;
  if (e >= EL) return;
  long long d = (e < Ee) ? ei[(size_t)Ee + e] : (long long)(e - Ee);
  float ee = __expf(elog[e] - emax[d]);
  elog[e] = ee;
  atomicAdd(&denom[d], ee);
}

// wave per edge: acc[dst] += (ee/denom[dst]) * xl[src]
template<int F>
__global__ void edge_acc_kernel(const float* __restrict__ xl,
                                const long long* __restrict__ ei, int Ee, int EL,
                                const float* __restrict__ elog,
                                const float* __restrict__ denom,
                                float* __restrict__ acc) {
  const int wid  = blockIdx.x * (blockDim.x >> 5) + (threadIdx.x >> 5);
  const int lane = threadIdx.x & 31;
  if (wid >= EL) return;
  long long s, d;
  if (wid < Ee) { s = ei[wid]; d = ei[(size_t)Ee + wid]; }
  else          { s = wid - Ee; d = s; }
  const float alpha = elog[wid] / denom[d];
  for (int f = lane; f < F; f += 32)
    atomicAdd(&acc[(size_t)d * F + f], alpha * xl[(size_t)s * F + f]);
}

__global__ void relu_kernel(float* a, long long n) {
  long long i = (long long)blockIdx.x * blockDim.x + threadIdx.x;
  if (i < n) a[i] = fmaxf(a[i], 0.0f);
}

// wave per node, C = 40: out = x - max - log(sum exp(x - max))
__global__ void log_softmax_kernel(const float* __restrict__ in,
                                   float* __restrict__ out, int nrows) {
  const int node = blockIdx.x * (blockDim.x >> 5) + (threadIdx.x >> 5);
  const int lane = threadIdx.x & 31;
  if (node >= nrows) return;
  const float* r = in + (size_t)node * 40;
  float v0 = r[lane];                                        // cols 0..31
  float v1 = (lane < 8) ? r[32 + lane] : -__builtin_inff();  // cols 32..39
  float mx = fmaxf(v0, v1);
  #pragma unroll
  for (int off = 16; off > 0; off >>= 1) mx = fmaxf(mx, __shfl_xor(mx, off, 32));
  float s = __expf(v0 - mx) + ((lane < 8) ? __expf(v1 - mx) : 0.0f);
  #pragma unroll
  for (int off = 16; off > 0; off >>= 1) s += __shfl_xor(s, off, 32);
  float lse = __logf(s);
  float* o = out + (size_t)node * 40;
  o[lane] = v0 - mx - lse;
  if (lane < 8) o[32 + lane] = v1 - mx - lse;
}

// ---------------------------------------------------------------------------
extern "C" void kernel_launch(void* const* d_in, const int* in_sizes, int n_in,
                              void* d_out, int out_size, void* d_ws, size_t ws_size,
                              hipStream_t stream) {
  const float*     x     = (const float*)d_in[0];
  const long long* ei    = (const long long*)d_in[1];
  const float* W1l   = (const float*)d_in[2];
  const float* b1l   = (const float*)d_in[3];
  const float* W1r   = (const float*)d_in[4];
  const float* b1r   = (const float*)d_in[5];
  const float* att1  = (const float*)d_in[6];
  const float* bias1 = (const float*)d_in[7];
  const float* lin1W = (const float*)d_in[8];
  const float* lin1b = (const float*)d_in[9];
  const float* W2l   = (const float*)d_in[10];
  const float* b2l   = (const float*)d_in[11];
  const float* W2r   = (const float*)d_in[12];
  const float* b2r   = (const float*)d_in[13];
  const float* att2  = (const float*)d_in[14];
  const float* bias2 = (const float*)d_in[15];
  const float* lin2W = (const float*)d_in[16];
  const float* lin2b = (const float*)d_in[17];

  const int FIN = 256, H = 64, C = 40;
  const int Nn = in_sizes[0] / FIN;
  const int Ee = in_sizes[1] / 2;
  const int EL = Ee + Nn;                 // with self loops

  // workspace carve-out (~66 MB)
  float* p = (float*)d_ws;
  float* xl1   = p; p += (size_t)Nn * H;
  float* xr1   = p; p += (size_t)Nn * H;
  float* acc1  = p; p += (size_t)Nn * H;  // lin1+bias1 seed, then += attn, then relu -> h
  float* xl2   = p; p += (size_t)Nn * C;
  float* xr2   = p; p += (size_t)Nn * C;
  float* acc2  = p; p += (size_t)Nn * C;  // lin2+bias2 seed, then += attn
  float* elog  = p; p += (size_t)EL;
  float* emax  = p; p += (size_t)Nn;
  float* denom = p; p += (size_t)Nn;

  // -------- layer 1 dense (WMMA): xl1, xr1, acc1 = lin1(x)+lin1_b+bias1
  {
    GemmTask t0{W1l, b1l, nullptr, xl1};
    GemmTask t1{W1r, b1r, nullptr, xr1};
    GemmTask t2{lin1W, lin1b, bias1, acc1};
    dim3 grid((Nn + 127) / 128, 3), blk(256);
    gemm_wmma_kernel<256, 64, 4><<<grid, blk, 0, stream>>>(x, Nn, t0, t1, t2);
  }
  seg_init_kernel<<<(Nn + 255) / 256, 256, 0, stream>>>(emax, denom, Nn);
  edge_logit_kernel<64><<<dim3((EL + 7) / 8), 256, 0, stream>>>(xl1, xr1, att1, ei, Ee, EL, elog, emax);
  edge_exp_kernel<<<(EL + 255) / 256, 256, 0, stream>>>(ei, Ee, EL, elog, emax, denom);
  edge_acc_kernel<64><<<dim3((EL + 7) / 8), 256, 0, stream>>>(xl1, ei, Ee, EL, elog, denom, acc1);
  relu_kernel<<<(unsigned)(((long long)Nn * H + 255) / 256), 256, 0, stream>>>(acc1, (long long)Nn * H);

  // -------- layer 2 dense (WMMA) on h = acc1
  {
    GemmTask t0{W2l, b2l, nullptr, xl2};
    GemmTask t1{W2r, b2r, nullptr, xr2};
    GemmTask t2{lin2W, lin2b, bias2, acc2};
    dim3 grid((Nn + 127) / 128, 3), blk(256);
    gemm_wmma_kernel<64, 40, 3><<<grid, blk, 0, stream>>>(acc1, Nn, t0, t1, t2);
  }
  seg_init_kernel<<<(Nn + 255) / 256, 256, 0, stream>>>(emax, denom, Nn);
  edge_logit_kernel<40><<<dim3((EL + 7) / 8), 256, 0, stream>>>(xl2, xr2, att2, ei, Ee, EL, elog, emax);
  edge_exp_kernel<<<(EL + 255) / 256, 256, 0, stream>>>(ei, Ee, EL, elog, emax, denom);
  edge_acc_kernel<40><<<dim3((EL + 7) / 8), 256, 0, stream>>>(xl2, ei, Ee, EL, elog, denom, acc2);

  // -------- epilogue: log_softmax -> d_out[0 : N*C), then edge_index tail
  log_softmax_kernel<<<dim3((Nn + 7) / 8), 256, 0, stream>>>(acc2, (float*)d_out, Nn);
  hipMemcpyAsync((char*)d_out + (size_t)Nn * C * sizeof(float), (const void*)ei,
                 (size_t)2 * Ee * sizeof(long long), hipMemcpyDeviceToDevice, stream);
}